// LengthRegulator_34600256537457
// MI455X (gfx1250) — compile-verified
//
#include <hip/hip_runtime.h>

// LengthRegulator (FastSpeech) for MI455X / gfx1250.
// B=32, T=512, D=384, TOUT=4096. Pure data movement:
//   HBM traffic ~ 192 MiB NT stores + ~24 MiB reads (xs stays L2-resident) -> ~10us @ 23.3 TB/s.
// No matmul -> no WMMA; CDNA5 paths: async global->LDS staging (ASYNCcnt),
// s_wait_asynccnt, global_prefetch_b8, non-temporal output stores.

#define LR_B    32
#define LR_T    512
#define LR_D    384
#define LR_TOUT 4096

typedef float fvec4 __attribute__((ext_vector_type(4)));

// ---------------------------------------------------------------------------
// Kernel 1: masked inclusive scan of durations, one block per batch.
//   cum[b,t] = sum_{i<=t} (i < ilens[b] ? ds[b,i] : 0)
// ---------------------------------------------------------------------------
__global__ __launch_bounds__(LR_T) void lr_scan_kernel(const int* __restrict__ ds,
                                                       const int* __restrict__ ilens,
                                                       int* __restrict__ cum) {
  const int b = blockIdx.x;
  const int t = threadIdx.x;
  __shared__ int s[LR_T];

  const int il = ilens[b];
  s[t] = (t < il) ? ds[b * LR_T + t] : 0;
  __syncthreads();

  // Hillis-Steele inclusive scan over 512 elements (9 steps).
  #pragma unroll
  for (int off = 1; off < LR_T; off <<= 1) {
    const int u = (t >= off) ? s[t - off] : 0;
    __syncthreads();
    s[t] += u;
    __syncthreads();
  }
  cum[b * LR_T + t] = s[t];
}

// ---------------------------------------------------------------------------
// Kernel 2: expand. Block = 256 threads (8 wave32 waves) handles 32 output
// rows of one batch. cum[] table staged into LDS via CDNA5 async loads.
// ---------------------------------------------------------------------------
__global__ __launch_bounds__(256) void lr_expand_kernel(const float* __restrict__ xs,
                                                        const int* __restrict__ cum,
                                                        float* __restrict__ out) {
  const int b       = blockIdx.y;
  const int rowBase = blockIdx.x * 32;
  const int wave    = threadIdx.x >> 5;
  const int lane    = threadIdx.x & 31;

  __shared__ int s_cum[LR_T];

  // --- CDNA5: asynchronous global->LDS staging of the cum table (2 KiB). ---
  {
    const int* g = cum + b * LR_T;
    for (int i = threadIdx.x; i < LR_T; i += 256) {
      const unsigned           lds_off = (unsigned)(size_t)(&s_cum[i]);
      const unsigned long long gaddr   = (unsigned long long)(size_t)(g + i);
      asm volatile("global_load_async_to_lds_b32 %0, %1, off"
                   :: "v"(lds_off), "v"(gaddr) : "memory");
    }
    asm volatile("s_wait_asynccnt 0" ::: "memory");
  }
  __syncthreads();

  const int total = s_cum[LR_T - 1];

  #pragma unroll
  for (int rr = 0; rr < 32; rr += 8) {
    const int t = rowBase + rr + wave;

    // searchsorted(cum, t, side='right'): first i with cum[i] > t.
    int lo = 0, hi = LR_T;
    while (lo < hi) {
      const int mid = (lo + hi) >> 1;
      if (s_cum[mid] <= t) lo = mid + 1; else hi = mid;
    }
    const int  idx   = (lo < LR_T - 1) ? lo : (LR_T - 1);  // clamp -> loads in-bounds
    const bool valid = (t < total);

    const fvec4* __restrict__ src =
        (const fvec4*)(xs + ((size_t)b * LR_T + idx) * LR_D);
    fvec4* __restrict__ dst =
        (fvec4*)(out + ((size_t)b * LR_TOUT + t) * LR_D);

    // CDNA5 prefetch of the gathered source row (global_prefetch_b8).
    __builtin_prefetch(src + lane, 0, 0);

    const fvec4 zero = {0.f, 0.f, 0.f, 0.f};
    // 384 floats = 96 fvec4; 32 lanes x 3 fvec4 each, stride-32 (coalesced B128).
    // Output is write-once (192 MiB > L2): use non-temporal stores so the
    // gathered xs rows (24 MiB) stay resident in the 192 MB L2.
    #pragma unroll
    for (int k = 0; k < 3; ++k) {
      const int c = lane + 32 * k;
      fvec4 v = src[c];
      v = valid ? v : zero;  // PAD via v_cndmask, no divergence
      __builtin_nontemporal_store(v, dst + c);
    }
  }
}

// ---------------------------------------------------------------------------
extern "C" void kernel_launch(void* const* d_in, const int* in_sizes, int n_in,
                              void* d_out, int out_size, void* d_ws, size_t ws_size,
                              hipStream_t stream) {
  const float* xs    = (const float*)d_in[0];  // (32, 512, 384) f32
  const int*   ds    = (const int*)d_in[1];    // (32, 512) int
  const int*   ilens = (const int*)d_in[2];    // (32,) int
  float*       out   = (float*)d_out;          // (32, 4096, 384) f32
  int*         cumws = (int*)d_ws;             // 64 KiB scratch: cum per (b,t)

  lr_scan_kernel<<<dim3(LR_B), dim3(LR_T), 0, stream>>>(ds, ilens, cumws);
  lr_expand_kernel<<<dim3(LR_TOUT / 32, LR_B), dim3(256), 0, stream>>>(xs, cumws, out);
}